// GIN_40475771797955
// MI455X (gfx1250) — compile-verified
//
#include <hip/hip_runtime.h>

#define NN 50000
#define NE 800000
#define D  64
#define D2 128

typedef __attribute__((ext_vector_type(2))) float v2f;
typedef __attribute__((ext_vector_type(8))) float v8f;

// ---------------- utility kernels ----------------

__global__ void zero_kernel(float* __restrict__ p, int n) {
  int i = blockIdx.x * blockDim.x + threadIdx.x;
  if (i < n) p[i] = 0.f;
}

__global__ void init_h_kernel(const int* __restrict__ x, const float* __restrict__ emb,
                              float* __restrict__ h) {
  int i = blockIdx.x * blockDim.x + threadIdx.x;
  if (i < NN * D) {
    int n = i >> 6, c = i & 63;
    h[i] = emb[x[n] * D + c];
  }
}

__global__ void deg_kernel(const int* __restrict__ ei, float* __restrict__ deg) {
  int i = blockIdx.x * blockDim.x + threadIdx.x;
  if (i < NE) atomicAdd(&deg[ei[NE + i]], 1.0f);
}

__global__ void invdeg_kernel(float* __restrict__ deg) {
  int i = blockIdx.x * blockDim.x + threadIdx.x;
  if (i < NN) deg[i] = 1.0f / fmaxf(deg[i], 1.0f);
}

// ---------------- edge scatter: agg[dst] += h[src] + (edge_attr @ eW + eb) ----
// grid-stride wave32-per-edge; lane j handles columns j and j+32.
// Edge-MLP weights are hoisted into registers once per wave; edge_attr read as 2x b128.
__global__ void scatter_kernel(const float* __restrict__ h, const int* __restrict__ ei,
                               const float* __restrict__ eattr, const float* __restrict__ eW,
                               const float* __restrict__ eb, float* __restrict__ agg) {
  int lane = threadIdx.x & 31;
  int wave = (blockIdx.x * blockDim.x + threadIdx.x) >> 5;
  int nwaves = (gridDim.x * blockDim.x) >> 5;

  float w0[8], w1[8];
#pragma unroll
  for (int k = 0; k < 8; k++) {
    w0[k] = eW[k * D + lane];
    w1[k] = eW[k * D + lane + 32];
  }
  float bias0 = eb[lane];
  float bias1 = eb[lane + 32];

  for (int e = wave; e < NE; e += nwaves) {
    int s = ei[e];
    int d = ei[NE + e];
    const float4* ea4 = (const float4*)(eattr + e * 8);
    float4 lo = ea4[0];
    float4 hi = ea4[1];
    float e0 = bias0, e1 = bias1;
    e0 = fmaf(lo.x, w0[0], e0); e1 = fmaf(lo.x, w1[0], e1);
    e0 = fmaf(lo.y, w0[1], e0); e1 = fmaf(lo.y, w1[1], e1);
    e0 = fmaf(lo.z, w0[2], e0); e1 = fmaf(lo.z, w1[2], e1);
    e0 = fmaf(lo.w, w0[3], e0); e1 = fmaf(lo.w, w1[3], e1);
    e0 = fmaf(hi.x, w0[4], e0); e1 = fmaf(hi.x, w1[4], e1);
    e0 = fmaf(hi.y, w0[5], e0); e1 = fmaf(hi.y, w1[5], e1);
    e0 = fmaf(hi.z, w0[6], e0); e1 = fmaf(hi.z, w1[6], e1);
    e0 = fmaf(hi.w, w0[7], e0); e1 = fmaf(hi.w, w1[7], e1);
    float m0 = h[s * D + lane] + e0;
    float m1 = h[s * D + 32 + lane] + e1;
    atomicAdd(&agg[d * D + lane], m0);
    atomicAdd(&agg[d * D + 32 + lane], m1);
  }
}

// ---------------- GEMM1: t = ((1+eps)h + agg*invdeg) @ W1 + b1, fused BN stats ----
// block = 128 threads = 4 waves; each wave owns a 16-row M tile; K=64, N=128.
__global__ void __launch_bounds__(128) gemm1_kernel(
    const float* __restrict__ h, const float* __restrict__ agg, const float* __restrict__ invdeg,
    const float* __restrict__ epsp, const float* __restrict__ W1, const float* __restrict__ b1,
    float* __restrict__ t, float* __restrict__ sum1, float* __restrict__ sq1) {
  __shared__ float sA[64 * 65];  // 64 rows x 64 cols, stride 65 (bank-conflict-free frag reads)
  int tid = threadIdx.x;
  int rowBase = blockIdx.x * 64;
  float eps1 = 1.0f + epsp[0];
  for (int i = tid; i < 64 * 64; i += 128) {
    int r = i >> 6, c = i & 63;
    int row = rowBase + r;
    float v = 0.f;
    if (row < NN) v = eps1 * h[row * D + c] + agg[row * D + c] * invdeg[row];
    sA[r * 65 + c] = v;
  }
  __syncthreads();
  int wave = tid >> 5, lane = tid & 31;
  int tileRow = rowBase + wave * 16;
  if (tileRow >= NN) return;  // wave-uniform: EXEC stays all-ones for WMMA

  v8f acc[8];
#pragma unroll
  for (int n = 0; n < 8; n++)
#pragma unroll
    for (int r = 0; r < 8; r++) acc[n][r] = 0.f;

  int aRow  = wave * 16 + (lane & 15);   // A frag: lane -> M = lane%16
  int kOff  = (lane >> 4) << 1;          // lanes 16-31 carry K+2,K+3
  int colLo = lane & 15;                 // B/C frag: lane -> N = lane%16

  for (int kc = 0; kc < 16; kc++) {
    int k = kc * 4 + kOff;
    v2f a;
    a.x = sA[aRow * 65 + k];
    a.y = sA[aRow * 65 + k + 1];
#pragma unroll
    for (int n = 0; n < 8; n++) {
      int col = n * 16 + colLo;
      v2f b;
      b.x = W1[k * D2 + col];
      b.y = W1[(k + 1) * D2 + col];
      acc[n] = __builtin_amdgcn_wmma_f32_16x16x4_f32(false, a, false, b, (short)0, acc[n],
                                                     false, false);
    }
  }

  int rOff = (lane >> 4) << 3;  // C frag: VGPR r -> row r + 8*(lane>=16)
  for (int n = 0; n < 8; n++) {
    int col = n * 16 + colLo;
    float bv = b1[col];
    float s = 0.f, q = 0.f;
#pragma unroll
    for (int r = 0; r < 8; r++) {
      float v = acc[n][r] + bv;
      t[(tileRow + rOff + r) * D2 + col] = v;
      s += v;
      q += v * v;
    }
    s += __shfl_xor(s, 16, 32);
    q += __shfl_xor(q, 16, 32);
    if (lane < 16) {
      atomicAdd(&sum1[col], s);
      atomicAdd(&sq1[col], q);
    }
  }
}

// ---------------- BN stat finalize: mu, rstd (ddof=0) ----------------
__global__ void bn_finalize_kernel(const float* __restrict__ sum, const float* __restrict__ sq,
                                   float* __restrict__ mu, float* __restrict__ rs, int nch) {
  int i = blockIdx.x * blockDim.x + threadIdx.x;
  if (i < nch) {
    float m = sum[i] * (1.0f / NN);
    float v = sq[i] * (1.0f / NN) - m * m;
    mu[i] = m;
    rs[i] = rsqrtf(v + 1e-5f);
  }
}

// ---------------- GEMM2: u = relu(BN1(t)) @ W2 + b2, fused BN2 stats ----
__global__ void __launch_bounds__(128) gemm2_kernel(
    const float* __restrict__ t, const float* __restrict__ mu1, const float* __restrict__ rs1,
    const float* __restrict__ g1, const float* __restrict__ be1,
    const float* __restrict__ W2, const float* __restrict__ b2,
    float* __restrict__ u, float* __restrict__ sum2, float* __restrict__ sq2) {
  __shared__ float sA[64 * 129];  // 64 rows x 128 cols, stride 129
  int tid = threadIdx.x;
  int rowBase = blockIdx.x * 64;
  for (int i = tid; i < 64 * 128; i += 128) {
    int r = i >> 7, c = i & 127;
    int row = rowBase + r;
    float v = 0.f;
    if (row < NN) {
      float tv = t[row * D2 + c];
      v = fmaxf(fmaf(g1[c] * (tv - mu1[c]), rs1[c], be1[c]), 0.f);
    }
    sA[r * 129 + c] = v;
  }
  __syncthreads();
  int wave = tid >> 5, lane = tid & 31;
  int tileRow = rowBase + wave * 16;
  if (tileRow >= NN) return;

  v8f acc[4];
#pragma unroll
  for (int n = 0; n < 4; n++)
#pragma unroll
    for (int r = 0; r < 8; r++) acc[n][r] = 0.f;

  int aRow  = wave * 16 + (lane & 15);
  int kOff  = (lane >> 4) << 1;
  int colLo = lane & 15;

  for (int kc = 0; kc < 32; kc++) {
    int k = kc * 4 + kOff;
    v2f a;
    a.x = sA[aRow * 129 + k];
    a.y = sA[aRow * 129 + k + 1];
#pragma unroll
    for (int n = 0; n < 4; n++) {
      int col = n * 16 + colLo;
      v2f b;
      b.x = W2[k * D + col];
      b.y = W2[(k + 1) * D + col];
      acc[n] = __builtin_amdgcn_wmma_f32_16x16x4_f32(false, a, false, b, (short)0, acc[n],
                                                     false, false);
    }
  }

  int rOff = (lane >> 4) << 3;
  for (int n = 0; n < 4; n++) {
    int col = n * 16 + colLo;
    float bv = b2[col];
    float s = 0.f, q = 0.f;
#pragma unroll
    for (int r = 0; r < 8; r++) {
      float v = acc[n][r] + bv;
      u[(tileRow + rOff + r) * D + col] = v;
      s += v;
      q += v * v;
    }
    s += __shfl_xor(s, 16, 32);
    q += __shfl_xor(q, 16, 32);
    if (lane < 16) {
      atomicAdd(&sum2[col], s);
      atomicAdd(&sq2[col], q);
    }
  }
}

// ---------------- h = relu(BN2(u)) ----------------
__global__ void bn2_relu_kernel(const float* __restrict__ u, const float* __restrict__ mu2,
                                const float* __restrict__ rs2, const float* __restrict__ g2,
                                const float* __restrict__ be2, float* __restrict__ h) {
  int i = blockIdx.x * blockDim.x + threadIdx.x;
  if (i < NN * D) {
    int c = i & 63;
    h[i] = fmaxf(fmaf(g2[c] * (u[i] - mu2[c]), rs2[c], be2[c]), 0.f);
  }
}

// ---------------- out = h @ pred_W + pred_b ; one wave per node ----------------
__global__ void pred_kernel(const float* __restrict__ h, const float* __restrict__ pW,
                            const float* __restrict__ pb, float* __restrict__ out) {
  int g = blockIdx.x * blockDim.x + threadIdx.x;
  int node = g >> 5, lane = g & 31;
  if (node >= NN) return;
  float x0 = h[node * D + lane];
  float x1 = h[node * D + 32 + lane];
  float a0 = x0 * pW[lane * 2 + 0] + x1 * pW[(lane + 32) * 2 + 0];
  float a1 = x0 * pW[lane * 2 + 1] + x1 * pW[(lane + 32) * 2 + 1];
#pragma unroll
  for (int off = 16; off > 0; off >>= 1) {
    a0 += __shfl_xor(a0, off, 32);
    a1 += __shfl_xor(a1, off, 32);
  }
  if (lane == 0) {
    out[node * 2 + 0] = a0 + pb[0];
    out[node * 2 + 1] = a1 + pb[1];
  }
}

// ---------------- driver ----------------

extern "C" void kernel_launch(void* const* d_in, const int* in_sizes, int n_in,
                              void* d_out, int out_size, void* d_ws, size_t ws_size,
                              hipStream_t stream) {
  const int*   x     = (const int*)d_in[0];
  const int*   ei    = (const int*)d_in[1];
  const float* eattr = (const float*)d_in[2];
  const float* emb   = (const float*)d_in[3];
  const float* eW    = (const float*)d_in[4];
  const float* eb    = (const float*)d_in[5];
  const float* pW    = (const float*)d_in[24];
  const float* pb    = (const float*)d_in[25];
  float* out = (float*)d_out;

  float* ws   = (float*)d_ws;
  float* h    = ws;              // NN*D
  float* agg  = h + NN * D;      // NN*D
  float* t    = agg + NN * D;    // NN*D2
  float* u    = t + NN * D2;     // NN*D
  float* deg  = u + NN * D;      // NN  (becomes inv_deg in place)
  float* stats = deg + NN;       // sum1[128] sq1[128] sum2[64] sq2[64] mu1[128] rs1[128] mu2[64] rs2[64]
  float* sum1 = stats;
  float* sq1  = sum1 + 128;
  float* sum2 = sq1 + 128;
  float* sq2  = sum2 + 64;
  float* mu1  = sq2 + 64;
  float* rs1  = mu1 + 128;
  float* mu2  = rs1 + 128;
  float* rs2  = mu2 + 64;

  const int T = 256;
  init_h_kernel<<<(NN * D + T - 1) / T, T, 0, stream>>>(x, emb, h);
  zero_kernel<<<(NN + T - 1) / T, T, 0, stream>>>(deg, NN);
  deg_kernel<<<(NE + T - 1) / T, T, 0, stream>>>(ei, deg);
  invdeg_kernel<<<(NN + T - 1) / T, T, 0, stream>>>(deg);

  int gemmBlocks = (NN + 63) / 64;  // 64 rows (4 M-tiles) per block
  int scatterBlocks = 4096;         // grid-stride: 32768 waves, ~24 edges each

  for (int l = 0; l < 2; l++) {
    int base = 6 + l * 9;
    const float* epsp = (const float*)d_in[base + 0];
    const float* W1   = (const float*)d_in[base + 1];
    const float* b1   = (const float*)d_in[base + 2];
    const float* g1   = (const float*)d_in[base + 3];
    const float* be1  = (const float*)d_in[base + 4];
    const float* W2   = (const float*)d_in[base + 5];
    const float* b2   = (const float*)d_in[base + 6];
    const float* g2   = (const float*)d_in[base + 7];
    const float* be2  = (const float*)d_in[base + 8];

    zero_kernel<<<(NN * D + T - 1) / T, T, 0, stream>>>(agg, NN * D);
    zero_kernel<<<2, T, 0, stream>>>(stats, 384);
    scatter_kernel<<<scatterBlocks, T, 0, stream>>>(h, ei, eattr, eW, eb, agg);
    gemm1_kernel<<<gemmBlocks, 128, 0, stream>>>(h, agg, deg, epsp, W1, b1, t, sum1, sq1);
    bn_finalize_kernel<<<1, 128, 0, stream>>>(sum1, sq1, mu1, rs1, 128);
    gemm2_kernel<<<gemmBlocks, 128, 0, stream>>>(t, mu1, rs1, g1, be1, W2, b2, u, sum2, sq2);
    bn_finalize_kernel<<<1, 64, 0, stream>>>(sum2, sq2, mu2, rs2, 64);
    bn2_relu_kernel<<<(NN * D + T - 1) / T, T, 0, stream>>>(u, mu2, rs2, g2, be2, h);
  }

  pred_kernel<<<(NN * 32 + T - 1) / T, T, 0, stream>>>(h, pW, pb, out);

  (void)in_sizes; (void)n_in; (void)out_size; (void)ws_size;
}